// Gaussian3D_69922067579281
// MI455X (gfx1250) — compile-verified
//
#include <hip/hip_runtime.h>
#include <hip/hip_bf16.h>

typedef __attribute__((ext_vector_type(2))) float v2f;
typedef __attribute__((ext_vector_type(8))) float v8f;

// Fixed problem shape (reference: M=8192 points, N=4096 gaussians, F=1).
constexpr int MM = 8192;
constexpr int NN = 4096;

// ---------------------------------------------------------------------------
// Kernel 0: per-Gaussian preprocessing.
//   A = Sigma^-1 = R diag(exp(-2*scale_raw)) R^T    (exact, no 3x3 inverse)
//   psi(k) = [A00,A11,A22, 2A01,2A02,2A12, -2(Ax)0,-2(Ax)1,-2(Ax)2, x^T A x, 0,0]
// Stored as k-PAIRS so the WMMA B-fragment becomes one b64 load per k-step:
//   psi2[(p*NN + n)*2 + {0,1}] = (psi(2p), psi(2p+1)),  p = 0..5
// Lane needs pair p = 2*step + hi  ->  lanes 0-15 read consecutive n: coalesced.
// ---------------------------------------------------------------------------
__global__ void gauss_prep(const float* __restrict__ xyz,
                           const float* __restrict__ scale_raw,
                           const float* __restrict__ rot_raw,
                           const float* __restrict__ op_raw,
                           float* __restrict__ psi2,  // 6*NN*2 floats
                           float* __restrict__ sig) { // NN floats
    int n = blockIdx.x * blockDim.x + threadIdx.x;
    if (n >= NN) return;

    float qw = rot_raw[n * 4 + 0], qx = rot_raw[n * 4 + 1];
    float qy = rot_raw[n * 4 + 2], qz = rot_raw[n * 4 + 3];
    float qn = rsqrtf(qw * qw + qx * qx + qy * qy + qz * qz);
    qw *= qn; qx *= qn; qy *= qn; qz *= qn;

    float R[3][3];
    R[0][0] = 1.f - 2.f * (qy * qy + qz * qz);
    R[0][1] = 2.f * (qx * qy - qw * qz);
    R[0][2] = 2.f * (qx * qz + qw * qy);
    R[1][0] = 2.f * (qx * qy + qw * qz);
    R[1][1] = 1.f - 2.f * (qx * qx + qz * qz);
    R[1][2] = 2.f * (qy * qz - qw * qx);
    R[2][0] = 2.f * (qx * qz - qw * qy);
    R[2][1] = 2.f * (qy * qz + qw * qx);
    R[2][2] = 1.f - 2.f * (qx * qx + qy * qy);

    float is2[3];
    is2[0] = __expf(-2.f * scale_raw[n * 3 + 0]);
    is2[1] = __expf(-2.f * scale_raw[n * 3 + 1]);
    is2[2] = __expf(-2.f * scale_raw[n * 3 + 2]);

    float A00 = 0.f, A01 = 0.f, A02 = 0.f, A11 = 0.f, A12 = 0.f, A22 = 0.f;
#pragma unroll
    for (int k = 0; k < 3; ++k) {
        A00 += R[0][k] * R[0][k] * is2[k];
        A01 += R[0][k] * R[1][k] * is2[k];
        A02 += R[0][k] * R[2][k] * is2[k];
        A11 += R[1][k] * R[1][k] * is2[k];
        A12 += R[1][k] * R[2][k] * is2[k];
        A22 += R[2][k] * R[2][k] * is2[k];
    }

    float x0 = xyz[n * 3 + 0], x1 = xyz[n * 3 + 1], x2 = xyz[n * 3 + 2];
    float Ax0 = A00 * x0 + A01 * x1 + A02 * x2;
    float Ax1 = A01 * x0 + A11 * x1 + A12 * x2;
    float Ax2 = A02 * x0 + A12 * x1 + A22 * x2;
    float cc  = x0 * Ax0 + x1 * Ax1 + x2 * Ax2;

    // pairs (k, k+1):  p0=(A00,A11) p1=(A22,2A01) p2=(2A02,2A12)
    //                  p3=(-2Ax0,-2Ax1) p4=(-2Ax2,c) p5=(0,0)
    psi2[(0 * NN + n) * 2 + 0] = A00;        psi2[(0 * NN + n) * 2 + 1] = A11;
    psi2[(1 * NN + n) * 2 + 0] = A22;        psi2[(1 * NN + n) * 2 + 1] = 2.f * A01;
    psi2[(2 * NN + n) * 2 + 0] = 2.f * A02;  psi2[(2 * NN + n) * 2 + 1] = 2.f * A12;
    psi2[(3 * NN + n) * 2 + 0] = -2.f * Ax0; psi2[(3 * NN + n) * 2 + 1] = -2.f * Ax1;
    psi2[(4 * NN + n) * 2 + 0] = -2.f * Ax2; psi2[(4 * NN + n) * 2 + 1] = cc;
    psi2[(5 * NN + n) * 2 + 0] = 0.f;        psi2[(5 * NN + n) * 2 + 1] = 0.f;

    sig[n] = 1.f / (1.f + __expf(-op_raw[n]));
}

// ---------------------------------------------------------------------------
// Kernel 1: WMMA GEMM (M x N x 12, f32) + exp + opacity + NT streaming stores.
// One block per 16-row strip of points; 8 waves stride over the 256 n-tiles.
// mahal tile = phi * psi via 3 chained V_WMMA_F32_16X16X4_F32 (k = 0..11).
// Stores: single base address + constant 16KB row-stride immediate offsets.
// values[] reduced deterministically (shfl tree -> LDS -> fixed-order sum).
// ---------------------------------------------------------------------------
__global__ __launch_bounds__(256)
void gauss_wmma(const float* __restrict__ points,  // MM x 3
                const float* __restrict__ psi2,    // 6 x NN x 2 (pair SoA)
                const float* __restrict__ sig,     // NN
                const float* __restrict__ feat,    // NN   (F == 1)
                float* __restrict__ values,        // MM
                float* __restrict__ weights) {     // MM x NN
    __shared__ float lds_vals[8][16];

    const int lane = threadIdx.x & 31;
    const int wid  = threadIdx.x >> 5;
    const int hi   = lane >> 4;      // 0: lanes 0-15, 1: lanes 16-31
    const int col  = lane & 15;
    const int m0   = blockIdx.x * 16;

    // A fragments: row m = m0 + col; VGPR0 -> K=2*hi, VGPR1 -> K=1+2*hi,
    // k-step base 4*s.  phi = [p0^2,p1^2,p2^2,p0p1,p0p2,p1p2,p0,p1,p2,1,0,0]
    const float p0 = points[(m0 + col) * 3 + 0];
    const float p1 = points[(m0 + col) * 3 + 1];
    const float p2 = points[(m0 + col) * 3 + 2];

    v2f a0, a1, a2;
    a0.x = hi ? (p2 * p2) : (p0 * p0);   // phi[2]  : phi[0]
    a0.y = hi ? (p0 * p1) : (p1 * p1);   // phi[3]  : phi[1]
    a1.x = hi ? p0        : (p0 * p2);   // phi[6]  : phi[4]
    a1.y = hi ? p1        : (p1 * p2);   // phi[7]  : phi[5]
    a2.x = hi ? 0.f       : p2;          // phi[10] : phi[8]
    a2.y = hi ? 0.f       : 1.f;         // phi[11] : phi[9]

    float acc[8];
#pragma unroll
    for (int r = 0; r < 8; ++r) acc[r] = 0.f;

    // Base row for this half-wave's D fragment rows (m0 + 8*hi + r).
    float* const wbase = weights + (size_t)(m0 + 8 * hi) * NN;

#pragma unroll 2
    for (int t = wid; t < (NN >> 4); t += 8) {
        const int n  = (t << 4) + col;
        const float sg = sig[n];
        const float fv = feat[n];

        // B fragments: pair p = 2*step + hi, one b64 load each (coalesced).
        const v2f b0 = *reinterpret_cast<const v2f*>(psi2 + 2 * ((0 + hi) * NN + n));
        const v2f b1 = *reinterpret_cast<const v2f*>(psi2 + 2 * ((2 + hi) * NN + n));
        const v2f b2 = *reinterpret_cast<const v2f*>(psi2 + 2 * ((4 + hi) * NN + n));

        v8f c = {0.f, 0.f, 0.f, 0.f, 0.f, 0.f, 0.f, 0.f};
        c = __builtin_amdgcn_wmma_f32_16x16x4_f32(false, a0, false, b0, (short)0, c, false, false);
        c = __builtin_amdgcn_wmma_f32_16x16x4_f32(false, a1, false, b1, (short)0, c, false, false);
        c = __builtin_amdgcn_wmma_f32_16x16x4_f32(false, a2, false, b2, (short)0, c, false, false);

        // D layout: VGPR r, lane L -> row m0 + 8*hi + r, col n.
        float* const wp = wbase + n;      // + r*NN folds to immediate offsets
#pragma unroll
        for (int r = 0; r < 8; ++r) {
            const float w = __expf(-0.5f * c[r]) * sg;
            __builtin_nontemporal_store(w, wp + r * NN);
            acc[r] += w * fv;
        }
    }

    // values: reduce across the 16 lanes of each half-wave, then across waves.
#pragma unroll
    for (int r = 0; r < 8; ++r) {
        float v = acc[r];
        v += __shfl_xor(v, 1, 16);
        v += __shfl_xor(v, 2, 16);
        v += __shfl_xor(v, 4, 16);
        v += __shfl_xor(v, 8, 16);
        if (lane == 0)  lds_vals[wid][r]     = v;   // rows m0+0..7
        if (lane == 16) lds_vals[wid][8 + r] = v;   // rows m0+8..15
    }
    __syncthreads();
    if (threadIdx.x < 16) {
        float s = 0.f;
#pragma unroll
        for (int w = 0; w < 8; ++w) s += lds_vals[w][threadIdx.x];
        values[m0 + threadIdx.x] = s;
    }
}

// ---------------------------------------------------------------------------
// Inputs (setup_inputs order): points(M*3), xyz(N*3), scale_raw(N*3),
// rotation_raw(N*4), opacity_raw(N*1), features(N*F).  Output: values(M*F)
// followed by weights(M*N), all f32.
// ---------------------------------------------------------------------------
extern "C" void kernel_launch(void* const* d_in, const int* in_sizes, int n_in,
                              void* d_out, int out_size, void* d_ws, size_t ws_size,
                              hipStream_t stream) {
    const float* points    = (const float*)d_in[0];
    const float* xyz       = (const float*)d_in[1];
    const float* scale_raw = (const float*)d_in[2];
    const float* rot_raw   = (const float*)d_in[3];
    const float* op_raw    = (const float*)d_in[4];
    const float* feat      = (const float*)d_in[5];

    float* values  = (float*)d_out;
    float* weights = (float*)d_out + (size_t)MM;   // F == 1

    float* psi2 = (float*)d_ws;                    // 6*NN*2 floats
    float* sig  = psi2 + (size_t)6 * NN * 2;       // NN floats

    gauss_prep<<<(NN + 255) / 256, 256, 0, stream>>>(xyz, scale_raw, rot_raw,
                                                     op_raw, psi2, sig);
    gauss_wmma<<<MM / 16, 256, 0, stream>>>(points, psi2, sig, feat,
                                            values, weights);
}